// RASCA_36292473651431
// MI455X (gfx1250) — compile-verified
//
#include <hip/hip_runtime.h>
#include <hip/hip_bf16.h>

typedef __attribute__((ext_vector_type(2))) float v2f;
typedef __attribute__((ext_vector_type(8))) float v8f;

#define C_DIM 2048
#define B_DIM 64
#define R_DIM 128
#define HW    192
#define NBC   131072      // B*C
#define NX    25165824    // B*C*H*W

// ---------------------------------------------------------------------------
// Pass 1: part-pooling + global pooling. One wave per (b,c) pair.
// Elements 0..191 per (b,c); part0 = [0,72), part1 = [72,128), part2 = [128,184),
// elements [184,192) (row 23) only contribute to the global mean.
// All part boundaries are even -> clean float2 lanes (lane owns e = 2*lane + 64*i).
// ---------------------------------------------------------------------------
__global__ void __launch_bounds__(256) reduce_pool_kernel(const float* __restrict__ x,
                                                          float* __restrict__ pooled,
                                                          float* __restrict__ gpool)
{
    int wid  = (blockIdx.x * blockDim.x + threadIdx.x) >> 5;  // global (b,c) index
    int lane = threadIdx.x & 31;
    const float* base = x + (size_t)wid * HW;
    float2 v0 = *(const float2*)(base + 2 * lane);         // e in [0,64)
    float2 v1 = *(const float2*)(base + 64 + 2 * lane);    // e in [64,128)
    float2 v2 = *(const float2*)(base + 128 + 2 * lane);   // e in [128,192)
    float a0 = v0.x + v0.y, a1 = v1.x + v1.y, a2 = v2.x + v2.y;
    float s0 = a0 + ((lane < 4) ? a1 : 0.f);   // e<72 -> part0
    float s1 = (lane >= 4) ? a1 : 0.f;         // 72<=e<128 -> part1
    float s2 = (lane < 28) ? a2 : 0.f;         // 128<=e<184 -> part2
    float st = a0 + a1 + a2;
    #pragma unroll
    for (int off = 16; off >= 1; off >>= 1) {
        s0 += __shfl_xor(s0, off, 32);
        s1 += __shfl_xor(s1, off, 32);
        s2 += __shfl_xor(s2, off, 32);
        st += __shfl_xor(st, off, 32);
    }
    if (lane == 0) {
        pooled[wid]           = s0 * (1.f / 72.f);   // 9 rows * 8
        pooled[NBC + wid]     = s1 * (1.f / 56.f);   // 7 rows * 8
        pooled[2 * NBC + wid] = s2 * (1.f / 56.f);   // 7 rows * 8
        gpool[wid]            = st * (1.f / 192.f);
    }
}

// ---------------------------------------------------------------------------
// Generic batched GEMM with relu+bias on the WMMA f32 path.
// D[p][m][n] = relu( sum_k A[p][m][k] * B[p][n][k] + bias[p][n] )
// M = 64, N = 128, K = 2048 (both use sites have exactly this shape).
// One wave per 16x16 output tile; V_WMMA_F32_16X16X4_F32, K stepped by 4.
// A-frag: lane holds A[m=lane%16][k0 + 2*(lane/16) .. +1] -> contiguous float2.
// B-frag: lane holds B^T[k][n=lane%16] = Bmem[n][k] -> contiguous float2 too.
// ---------------------------------------------------------------------------
__global__ void __launch_bounds__(32) wmma_gemm_relu_kernel(const float* __restrict__ A,
                                                            const float* __restrict__ Bm,
                                                            const float* __restrict__ bias,
                                                            float* __restrict__ D,
                                                            long aS, long bS, long biasS, long dS)
{
    int p = blockIdx.z;
    const float* Ap   = A + p * aS;
    const float* Bp   = Bm + p * bS;
    const float* bip  = bias + p * biasS;
    float*       Dp   = D + p * dS;
    int lane = threadIdx.x;
    int r16  = lane & 15;
    int kh   = lane >> 4;           // selects K pair 0/1 (A) and rows j / j+8 (C/D)
    int m0 = blockIdx.y * 16, n0 = blockIdx.x * 16;
    v8f acc;
    #pragma unroll
    for (int i = 0; i < 8; i++) acc[i] = 0.f;
    const float* arow = Ap + (long)(m0 + r16) * C_DIM + 2 * kh;
    const float* brow = Bp + (long)(n0 + r16) * C_DIM + 2 * kh;
    for (int k0 = 0; k0 < C_DIM; k0 += 4) {
        v2f a = *(const v2f*)(arow + k0);
        v2f b = *(const v2f*)(brow + k0);
        acc = __builtin_amdgcn_wmma_f32_16x16x4_f32(false, a, false, b, (short)0, acc,
                                                    false, false);
    }
    #pragma unroll
    for (int j = 0; j < 8; j++) {
        int row = m0 + j + 8 * kh;
        int col = n0 + r16;
        float v = acc[j] + bip[col];
        Dp[row * R_DIM + col] = fmaxf(v, 0.f);
    }
}

// ---------------------------------------------------------------------------
// Part weights (softmax over 3 logits) + sparsity gate k[b]. One wave per b.
// ---------------------------------------------------------------------------
__global__ void __launch_bounds__(32) pw_sparsity_kernel(const float* __restrict__ gpool,
                                                         const float* __restrict__ t,
                                                         const float* __restrict__ pw_w,
                                                         const float* __restrict__ pw_b,
                                                         const float* __restrict__ sg_w2,
                                                         const float* __restrict__ sg_b2,
                                                         float* __restrict__ pw,
                                                         int* __restrict__ kArr,
                                                         float* __restrict__ pw_out)
{
    int b = blockIdx.x, lane = threadIdx.x;
    float l0 = 0.f, l1 = 0.f, l2 = 0.f, sp = 0.f;
    const float* g = gpool + b * C_DIM;
    for (int i = lane; i < C_DIM; i += 32) {
        float gv = g[i];
        l0 += gv * pw_w[i];
        l1 += gv * pw_w[C_DIM + i];
        l2 += gv * pw_w[2 * C_DIM + i];
    }
    for (int i = lane; i < R_DIM; i += 32) sp += t[b * R_DIM + i] * sg_w2[i];
    #pragma unroll
    for (int off = 16; off >= 1; off >>= 1) {
        l0 += __shfl_xor(l0, off, 32);
        l1 += __shfl_xor(l1, off, 32);
        l2 += __shfl_xor(l2, off, 32);
        sp += __shfl_xor(sp, off, 32);
    }
    if (lane == 0) {
        l0 += pw_b[0]; l1 += pw_b[1]; l2 += pw_b[2];
        float mx = fmaxf(l0, fmaxf(l1, l2));
        float e0 = __expf(l0 - mx), e1 = __expf(l1 - mx), e2 = __expf(l2 - mx);
        float inv = 1.f / (e0 + e1 + e2);
        float p0 = e0 * inv, p1 = e1 * inv, p2 = e2 * inv;
        pw[b * 3 + 0] = p0; pw[b * 3 + 1] = p1; pw[b * 3 + 2] = p2;
        pw_out[b * 3 + 0] = p0; pw_out[b * 3 + 1] = p1; pw_out[b * 3 + 2] = p2;
        float s = 1.f / (1.f + __expf(-(sp + sg_b2[0])));
        int k = (int)floorf(s * (float)C_DIM);
        k = min(max(k, 1), C_DIM);
        kArr[b] = k;
    }
}

// ---------------------------------------------------------------------------
// fused[b,c] = sum_p pw[b,p] * sigmoid( h1[p,b,:] . pa_w2[p,c,:] + pa_b2[p,c] )
// WMMA over K = R = 128, all 3 parts accumulated in-register (att never stored).
// ---------------------------------------------------------------------------
__global__ void __launch_bounds__(32) fused_wmma_kernel(const float* __restrict__ h1,
                                                        const float* __restrict__ w2,
                                                        const float* __restrict__ b2,
                                                        const float* __restrict__ pw,
                                                        float* __restrict__ fused)
{
    int lane = threadIdx.x;
    int r16 = lane & 15, kh = lane >> 4;
    int m0 = blockIdx.y * 16, n0 = blockIdx.x * 16;
    float facc[8];
    #pragma unroll
    for (int j = 0; j < 8; j++) facc[j] = 0.f;
    for (int p = 0; p < 3; p++) {
        const float* Ap = h1 + p * (B_DIM * R_DIM);
        const float* Bp = w2 + (long)p * (C_DIM * R_DIM);
        v8f acc;
        #pragma unroll
        for (int i = 0; i < 8; i++) acc[i] = 0.f;
        const float* arow = Ap + (m0 + r16) * R_DIM + 2 * kh;
        const float* brow = Bp + (long)(n0 + r16) * R_DIM + 2 * kh;
        #pragma unroll 8
        for (int k0 = 0; k0 < R_DIM; k0 += 4) {
            v2f a = *(const v2f*)(arow + k0);
            v2f b = *(const v2f*)(brow + k0);
            acc = __builtin_amdgcn_wmma_f32_16x16x4_f32(false, a, false, b, (short)0, acc,
                                                        false, false);
        }
        #pragma unroll
        for (int j = 0; j < 8; j++) {
            int bg = m0 + j + 8 * kh;
            int cg = n0 + r16;
            float v = acc[j] + b2[p * C_DIM + cg];
            float sig = 1.f / (1.f + __expf(-v));
            facc[j] += pw[bg * 3 + p] * sig;
        }
    }
    #pragma unroll
    for (int j = 0; j < 8; j++)
        fused[(m0 + j + 8 * kh) * C_DIM + n0 + r16] = facc[j];
}

// ---------------------------------------------------------------------------
// Exact top-k mask matching double-argsort semantics: rank = #{strictly greater}
// + #{equal at lower index}; mask = rank < k[b]. Fused row staged in LDS, 16
// blocks per b to spread the O(C^2) scan across WGPs. fa = fused * mask.
// ---------------------------------------------------------------------------
__global__ void __launch_bounds__(128) rank_mask_kernel(const float* __restrict__ fused,
                                                        const int* __restrict__ kArr,
                                                        float* __restrict__ fa)
{
    __shared__ float sf[C_DIM];
    int b = blockIdx.y, tid = threadIdx.x;
    for (int i = tid; i < C_DIM; i += 128) sf[i] = fused[b * C_DIM + i];
    __syncthreads();
    int c = blockIdx.x * 128 + tid;
    float fc = sf[c];
    int cnt = 0;
    for (int j = 0; j < C_DIM; j++) {
        float fv = sf[j];
        cnt += (fv > fc) || ((fv == fc) && (j < c));
    }
    fa[b * C_DIM + c] = (cnt < kArr[b]) ? fc : 0.f;
}

// ---------------------------------------------------------------------------
// Streaming apply: out1 = x * fa[b,c], out2 = x - out1. float4 vectorized;
// x re-read here hits the 192MB L2 (x is only ~100MB). Grid covers exactly
// B*C*HW/4 = 6,291,456 float4s.
// ---------------------------------------------------------------------------
__global__ void __launch_bounds__(256) apply_kernel(const float4* __restrict__ x4,
                                                    const float* __restrict__ fa,
                                                    float4* __restrict__ o1,
                                                    float4* __restrict__ o2)
{
    int idx = blockIdx.x * 256 + threadIdx.x;
    float4 v = x4[idx];
    float s = fa[idx / 48];          // 48 float4s per (b,c) region
    float4 a; a.x = v.x * s; a.y = v.y * s; a.z = v.z * s; a.w = v.w * s;
    float4 r; r.x = v.x - a.x; r.y = v.y - a.y; r.z = v.z - a.z; r.w = v.w - a.w;
    o1[idx] = a;
    o2[idx] = r;
}

extern "C" void kernel_launch(void* const* d_in, const int* in_sizes, int n_in,
                              void* d_out, int out_size, void* d_ws, size_t ws_size,
                              hipStream_t stream)
{
    const float* x     = (const float*)d_in[0];
    const float* pa_w1 = (const float*)d_in[1];
    const float* pa_b1 = (const float*)d_in[2];
    const float* pa_w2 = (const float*)d_in[3];
    const float* pa_b2 = (const float*)d_in[4];
    const float* pw_w  = (const float*)d_in[5];
    const float* pw_b  = (const float*)d_in[6];
    const float* sg_w1 = (const float*)d_in[7];
    const float* sg_b1 = (const float*)d_in[8];
    const float* sg_w2 = (const float*)d_in[9];
    const float* sg_b2 = (const float*)d_in[10];
    float* out = (float*)d_out;

    // workspace layout (floats): ~3.3 MB total
    float* ws     = (float*)d_ws;
    float* pooled = ws;                         // 3*NBC
    float* gpool  = pooled + 3 * NBC;           // NBC
    float* h1     = gpool + NBC;                // 3*B*R
    float* tb     = h1 + 3 * B_DIM * R_DIM;     // B*R
    float* pw     = tb + B_DIM * R_DIM;         // B*3 (padded to B*4)
    float* fusedv = pw + B_DIM * 4;             // NBC
    float* fav    = fusedv + NBC;               // NBC
    int*   kArr   = (int*)(fav + NBC);          // B

    // 1) pooling pass over x (one read of x)
    reduce_pool_kernel<<<16384, 256, 0, stream>>>(x, pooled, gpool);
    // 2) h1[p] = relu(pooled[p] . pa_w1[p]^T + pa_b1[p])   (WMMA, P=3)
    wmma_gemm_relu_kernel<<<dim3(8, 4, 3), 32, 0, stream>>>(
        pooled, pa_w1, pa_b1, h1,
        (long)NBC, (long)(R_DIM * C_DIM), (long)R_DIM, (long)(B_DIM * R_DIM));
    // 3) t = relu(gpool . sg_w1^T + sg_b1)                 (WMMA, P=1)
    wmma_gemm_relu_kernel<<<dim3(8, 4, 1), 32, 0, stream>>>(
        gpool, sg_w1, sg_b1, tb, 0L, 0L, 0L, 0L);
    // 4) part weights (softmax) + k[b]; also writes pw to output tail
    pw_sparsity_kernel<<<64, 32, 0, stream>>>(gpool, tb, pw_w, pw_b, sg_w2, sg_b2,
                                              pw, kArr, out + 2L * NX);
    // 5) fused[b,c] (WMMA over R, 3 parts fused with sigmoid + pw weighting)
    fused_wmma_kernel<<<dim3(128, 4), 32, 0, stream>>>(h1, pa_w2, pa_b2, pw, fusedv);
    // 6) exact top-k mask -> fa
    rank_mask_kernel<<<dim3(16, 64), 128, 0, stream>>>(fusedv, kArr, fav);
    // 7) streaming apply -> out1, out2
    apply_kernel<<<24576, 256, 0, stream>>>((const float4*)x, fav,
                                            (float4*)out, (float4*)(out + NX));
    (void)in_sizes; (void)n_in; (void)out_size; (void)ws_size;
}